// TailTransform_46703474377183
// MI455X (gfx1250) — compile-verified
//
#include <hip/hip_runtime.h>
#include <hip/hip_bf16.h>
#include <stdint.h>

// ---------------------------------------------------------------------------
// TailTransform: piecewise Student-t (nu=4 pos / nu=10 neg) -> Gaussianize.
// Elementwise, memory-bound target (768 MB @ 23.3 TB/s ~ 33us floor). CDNA5
// async global->LDS double-buffered pipeline + wave32 + B128 NT stores.
// Math uses raw v_rsq/v_sqrt/v_log/v_rcp to stay off the VALU ceiling.
// ---------------------------------------------------------------------------

#define THREADS 256
#define MAX_BLOCKS 4096

typedef float f4 __attribute__((ext_vector_type(4)));
typedef int   v4i __attribute__((vector_size(16)));   // matches builtin param type

#if defined(__AMDGCN__) && __has_builtin(__builtin_amdgcn_global_load_async_to_lds_b128) && __has_builtin(__builtin_amdgcn_s_wait_asynccnt)
#define USE_ASYNC_LDS 1
#else
#define USE_ASYNC_LDS 0
#endif

#if USE_ASYNC_LDS
typedef __attribute__((address_space(1))) v4i* gas_v4i_ptr;
typedef __attribute__((address_space(3))) v4i* las_v4i_ptr;
#endif

// ---- raw-rate transcendentals (1 VALU op each on gfx1250), guarded --------
#if defined(__AMDGCN__) && __has_builtin(__builtin_amdgcn_rsqf)
#define FAST_RSQ(x)  __builtin_amdgcn_rsqf(x)
#else
#define FAST_RSQ(x)  rsqrtf(x)
#endif

#if defined(__AMDGCN__) && __has_builtin(__builtin_amdgcn_sqrtf)
#define FAST_SQRT(x) __builtin_amdgcn_sqrtf(x)
#else
#define FAST_SQRT(x) __builtin_sqrtf(x)
#endif

#if defined(__AMDGCN__) && __has_builtin(__builtin_amdgcn_logf)
#define FAST_LOG2(x) __builtin_amdgcn_logf(x)       // v_log_f32: log2(x)
#else
#define FAST_LOG2(x) log2f(x)
#endif

#if defined(__AMDGCN__) && __has_builtin(__builtin_amdgcn_rcpf)
#define FAST_RCP(x)  __builtin_amdgcn_rcpf(x)
#else
#define FAST_RCP(x)  (1.0f / (x))
#endif

#define LN2F 0.69314718056f

__device__ __forceinline__ void async_load_f4(const f4* gsrc, f4* ldst) {
#if USE_ASYNC_LDS
  gas_v4i_ptr g = (gas_v4i_ptr)(uintptr_t)gsrc;
  las_v4i_ptr l = (las_v4i_ptr)(uint32_t)(uintptr_t)ldst;
  __builtin_amdgcn_global_load_async_to_lds_b128(g, l, 0, 0);
#else
  (void)gsrc; (void)ldst;
#endif
}

// ---- inverse normal CDF (Acklam), valid for p in (0, 0.5], returns x <= 0 --
__device__ __forceinline__ float ndtri_neg(float p) {
  const float AK1 = -3.969683028665376e+01f, AK2 = 2.209460984245205e+02f,
              AK3 = -2.759285104469687e+02f, AK4 = 1.383577518672690e+02f,
              AK5 = -3.066479806614716e+01f, AK6 = 2.506628277459239e+00f;
  const float BK1 = -5.447609879822406e+01f, BK2 = 1.615858368580409e+02f,
              BK3 = -1.556989798598866e+02f, BK4 = 6.680131188771972e+01f,
              BK5 = -1.328068155288572e+01f;
  const float CK1 = -7.784894002430293e-03f, CK2 = -3.223964580411365e-01f,
              CK3 = -2.400758277161838e+00f, CK4 = -2.549732539343734e+00f,
              CK5 = 4.374664141464968e+00f,  CK6 = 2.938163982698783e+00f;
  const float DK1 = 7.784695709041462e-03f,  DK2 = 3.224671290700398e-01f,
              DK3 = 2.445134137142996e+00f,  DK4 = 3.754408661907416e+00f;

  if (p >= 0.02425f) {
    float q = p - 0.5f;
    float r = q * q;
    float num = fmaf(r, fmaf(r, fmaf(r, fmaf(r, fmaf(r, AK1, AK2), AK3), AK4), AK5), AK6);
    float den = fmaf(r, fmaf(r, fmaf(r, fmaf(r, fmaf(r, BK1, BK2), BK3), BK4), BK5), 1.0f);
    return q * num * FAST_RCP(den);
  } else {
    // sqrt(-2 ln p) = sqrt( -2*ln2 * log2(p) )
    float t = FAST_SQRT(FAST_LOG2(p) * (-2.0f * LN2F));
    float num = fmaf(t, fmaf(t, fmaf(t, fmaf(t, fmaf(t, CK1, CK2), CK3), CK4), CK5), CK6);
    float den = fmaf(t, fmaf(t, fmaf(t, fmaf(t, DK1, DK2), DK3), DK4), 1.0f);
    return num * FAST_RCP(den);
  }
}

// ---- full per-element transform ------------------------------------------
struct UL { float u; float l; };

__device__ __forceinline__ UL tail_transform(float z) {
  const bool pos = z > 0.0f;
  const float s  = z * z;
  const float nu = pos ? 4.0f : 10.0f;

  // Closed-form Student-t upper-tail prob at t=|z| (even nu):
  // p = 1/2 - (t / (2*sqrt(t^2+nu))) * sum_j a_j * (nu/(t^2+nu))^j
  const float r = FAST_RSQ(s + nu);      // 1/sqrt(s+nu)
  const float q = nu * (r * r);          // nu/(s+nu) in (0,1]
  const float c = fabsf(z) * r;          // |z|/sqrt(s+nu)
  // a1=1/2 always; a2..a4 only for nu=10 (3/8, 5/16, 35/128)
  const float k2 = pos ? 0.0f : 0.375f;
  const float k3 = pos ? 0.0f : 0.3125f;
  const float k4 = pos ? 0.0f : 0.2734375f;
  const float poly = fmaf(q, fmaf(q, fmaf(q, fmaf(q, k4, k3), k2), 0.5f), 1.0f);
  float p = fmaf(-0.5f * c, poly, 0.5f); // tail prob, <= 0.5
  p = fmaxf(p, 1e-7f);                   // matches reference EPS clip

  // ndtri(1-p) = -ndtri(p)
  const float x = ndtri_neg(p);          // <= 0
  const float u = pos ? -x : x;

  // log f_t(z) = C_nu - (nu+1)/2 * ln2 * log2(1 + z^2/nu)
  const float Cn    = pos ? -0.98082925f : -0.94389735f; // lnG((nu+1)/2)-lnG(nu/2)-0.5*ln(nu*pi)
  const float hpln2 = pos ? 1.73286795f  : 3.81230949f;  // (nu+1)/2 * ln2
  const float invnu = pos ? 0.25f : 0.1f;
  const float log_ft = fmaf(-hpln2, FAST_LOG2(fmaf(s, invnu, 1.0f)), Cn);
  // lad = log_ft + 0.5*u^2 + 0.5*ln(2pi)
  UL out;
  out.u = u;
  out.l = fmaf(0.5f * u, u, log_ft + 0.91893853f);
  return out;
}

__device__ __forceinline__ void transform4(const f4 zv, f4& uv, f4& lv) {
  UL r0 = tail_transform(zv.x);
  UL r1 = tail_transform(zv.y);
  UL r2 = tail_transform(zv.z);
  UL r3 = tail_transform(zv.w);
  uv.x = r0.u; uv.y = r1.u; uv.z = r2.u; uv.w = r3.u;
  lv.x = r0.l; lv.y = r1.l; lv.z = r2.l; lv.w = r3.l;
}

__global__ __launch_bounds__(THREADS) void
TailTransform_kernel(const float* __restrict__ z,
                     float* __restrict__ u,
                     float* __restrict__ lad,
                     int n4, int n) {
  const int tid    = threadIdx.x;
  const int stride = gridDim.x * THREADS;
  const int idx0   = blockIdx.x * THREADS + tid;

  const f4* __restrict__ z4 = (const f4*)z;
  f4* __restrict__ u4 = (f4*)u;
  f4* __restrict__ l4 = (f4*)lad;

#if USE_ASYNC_LDS
  __shared__ f4 tile[2][THREADS];
  // Prologue: async-stage first tile into LDS (buffer 0).
  if (idx0 < n4) async_load_f4(z4 + idx0, &tile[0][tid]);
  int buf = 0;
  for (int i = idx0; i < n4; i += stride, buf ^= 1) {
    const int nxt = i + stride;
    const bool have_next = nxt < n4;
    if (have_next) {
      // Stage next tile while we compute the current one.
      async_load_f4(z4 + nxt, &tile[buf ^ 1][tid]);
      __builtin_amdgcn_s_wait_asynccnt(1);   // retire current tile (in-order)
    } else {
      __builtin_amdgcn_s_wait_asynccnt(0);
    }
    asm volatile("" ::: "memory");           // don't hoist LDS read above wait
    const f4 zv = tile[buf][tid];            // ds_load_b128
    f4 uv, lv;
    transform4(zv, uv, lv);
    // Streaming outputs: no reuse, keep them out of L2 (TH_NT).
    __builtin_nontemporal_store(uv, u4 + i);
    __builtin_nontemporal_store(lv, l4 + i);
  }
#else
  for (int i = idx0; i < n4; i += stride) {
    const f4 zv = z4[i];
    f4 uv, lv;
    transform4(zv, uv, lv);
    __builtin_nontemporal_store(uv, u4 + i);
    __builtin_nontemporal_store(lv, l4 + i);
  }
#endif

  // Scalar remainder (n not divisible by 4): handled by block 0.
  const int rem = n - (n4 << 2);
  if (blockIdx.x == 0 && tid < rem) {
    const int j = (n4 << 2) + tid;
    UL rr = tail_transform(z[j]);
    u[j] = rr.u;
    lad[j] = rr.l;
  }
}

extern "C" void kernel_launch(void* const* d_in, const int* in_sizes, int n_in,
                              void* d_out, int out_size, void* d_ws, size_t ws_size,
                              hipStream_t stream) {
  (void)n_in; (void)out_size; (void)d_ws; (void)ws_size;
  const float* z = (const float*)d_in[0];
  const int n  = in_sizes[0];
  const int n4 = n >> 2;
  float* u   = (float*)d_out;
  float* lad = (float*)d_out + n;

  int blocks = (n4 + THREADS - 1) / THREADS;
  if (blocks < 1) blocks = 1;
  if (blocks > MAX_BLOCKS) blocks = MAX_BLOCKS;

  TailTransform_kernel<<<blocks, THREADS, 0, stream>>>(z, u, lad, n4, n);
}